// RectNetCSPN_31619549233940
// MI455X (gfx1250) — compile-verified
//
#include <hip/hip_runtime.h>
#include <cstdint>

// CSPN: 16 steps of 8-way propagation. Each step:
//   out[c,p] = box3x3(|g[c]|*d)[p] / box3x3(|g[c]|)[p]   (zero padded)
//   d[p]     = sparse[p] > 0 ? sparse[p] : max_c out[c,p]
// Memory-bound, fully L2-resident on MI455X (88MB working set < 192MB L2).
// Halo tiles are streamed into LDS with CDNA5 async-to-LDS loads (b128 for
// the aligned interior, b32 for the two edge columns), single-buffered, prefetched one
// guidance channel ahead of the compute (ASYNCcnt + barrier pipeline).

namespace {
constexpr int H  = 512;
constexpr int W  = 1024;
constexpr int NB = 4;
constexpr int C  = 8;
constexpr int HWp = H * W;          // plane size
constexpr int T  = 32;              // tile edge
constexpr int HA = T + 2;           // halo tile edge (34)
constexpr int RSTR = 40;            // LDS row stride (floats): 160B, 16B-aligned
constexpr int XOFF = 3;             // left-halo column index; interior at XOFF+1
                                    // -> interior LDS byte off = row*160+16+16k
constexpr int NTHREADS = 256;       // 8 wave32s
}

// CDNA5 async global->LDS copies (tracked by ASYNCcnt).
__device__ __forceinline__ void async_ld_b32(float* lds_ptr, const float* gptr) {
    uint32_t loff  = (uint32_t)(uintptr_t)lds_ptr;   // low 32 bits = LDS offset
    uint64_t gaddr = (uint64_t)(uintptr_t)gptr;
    asm volatile("global_load_async_to_lds_b32 %0, %1, off"
                 :: "v"(loff), "v"(gaddr) : "memory");
}
__device__ __forceinline__ void async_ld_b128(float* lds_ptr, const float* gptr) {
    uint32_t loff  = (uint32_t)(uintptr_t)lds_ptr;   // must be 16B aligned
    uint64_t gaddr = (uint64_t)(uintptr_t)gptr;      // must be 16B aligned
    asm volatile("global_load_async_to_lds_b128 %0, %1, off"
                 :: "v"(loff), "v"(gaddr) : "memory");
}
__device__ __forceinline__ void wait_async0() {
    asm volatile("s_wait_asynccnt 0x0" ::: "memory");
}

// Issue async loads for one 34x34 halo tile from plane `src` into LDS `tile`.
// Work items: 34 rows x 10 slots (8 x b128 interior chunks + 2 edge b32).
// Rows/edges outside the image are skipped -> LDS keeps its zero fill.
__device__ __forceinline__ void load_halo_async(float* tile, const float* src,
                                                int tile_x, int tile_y, int tid) {
    for (int j = tid; j < HA * 10; j += NTHREADS) {
        int hy = j / 10, s = j - hy * 10;
        int gy = tile_y + hy - 1;
        if ((unsigned)gy >= (unsigned)H) continue;
        const float* row = src + gy * W;
        float* lrow = tile + hy * RSTR + XOFF;
        if (s < 8) {
            // interior: gx = tile_x + 4s .. +3, always in-bounds, 16B aligned
            async_ld_b128(lrow + 1 + 4 * s, row + tile_x + 4 * s);
        } else if (s == 8) {
            int gx = tile_x - 1;
            if (gx >= 0) async_ld_b32(lrow, row + gx);
        } else {
            int gx = tile_x + T;
            if (gx < W) async_ld_b32(lrow + 1 + T, row + gx);
        }
    }
}

// d0 = sparse>0 ? sparse : blur   (== (1-sign(s))*blur + sign(s)*s for s>=0)
__global__ __launch_bounds__(NTHREADS)
void cspn_init(const float4* __restrict__ blur,
               const float4* __restrict__ sparse,
               float4* __restrict__ dout, int n4) {
    int i = blockIdx.x * NTHREADS + threadIdx.x;
    if (i < n4) {
        float4 b = blur[i], s = sparse[i], o;
        o.x = (s.x > 0.f) ? s.x : b.x;
        o.y = (s.y > 0.f) ? s.y : b.y;
        o.z = (s.z > 0.f) ? s.z : b.z;
        o.w = (s.w > 0.f) ? s.w : b.w;
        dout[i] = o;
    }
}

__global__ __launch_bounds__(NTHREADS)
void cspn_step(const float* __restrict__ g,       // [B,8,H,W]
               const float* __restrict__ sparse,  // [B,1,H,W]
               const float* __restrict__ din,     // [B,1,H,W]
               float* __restrict__ dout) {        // [B,1,H,W]
    __shared__ float d_raw[HA * RSTR];  // depth halo tile (raw)
    __shared__ float g_raw[HA * RSTR];  // guidance halo tile (current channel)
    __shared__ float rs_g[HA * T];      // horizontal 3-sum of |g|
    __shared__ float rs_p[HA * T];      // horizontal 3-sum of |g|*d

    const int tid    = threadIdx.x;
    const int tile_x = blockIdx.x * T;
    const int tile_y = blockIdx.y * T;
    const int b      = blockIdx.z;

    const float* __restrict__ dinb = din    + b * HWp;
    const float* __restrict__ gb   = g      + b * (C * HWp);
    const float* __restrict__ spb  = sparse + b * HWp;
    float* __restrict__       outb = dout   + b * HWp;

    // ---- zero raw tiles once: out-of-image halo slots are never written by
    //      the async loads (EXEC-masked / row-skipped) -> zero padding for all
    //      8 channels comes for free.
    for (int i = tid; i < HA * RSTR; i += NTHREADS) {
        d_raw[i] = 0.f;
        g_raw[i] = 0.f;
    }
    __syncthreads();   // zeros ordered before any async write lands

    // ---- async-load d halo tile + channel-0 guidance halo tile
    load_halo_async(d_raw, dinb, tile_x, tile_y, tid);
    load_halo_async(g_raw, gb,   tile_x, tile_y, tid);
    wait_async0();
    __syncthreads();

    const int tx = tid & (T - 1);
    const int ty = tid >> 5;                   // 0..7, each thread owns 4 rows
    float maxv[4] = { -__builtin_inff(), -__builtin_inff(),
                      -__builtin_inff(), -__builtin_inff() };

    for (int c = 0; c < C; ++c) {
        // ---- phase A: horizontal 3-sums, 34 halo rows x 32 interior cols
        for (int i = tid; i < HA * T; i += NTHREADS) {
            int y = i >> 5, x = i & (T - 1);
            int p = y * RSTR + XOFF + x;       // halo cols x..x+2, center x+1
            float g0 = fabsf(g_raw[p]);
            float g1 = fabsf(g_raw[p + 1]);
            float g2 = fabsf(g_raw[p + 2]);
            rs_g[i] = g0 + g1 + g2;
            rs_p[i] = g0 * d_raw[p] + g1 * d_raw[p + 1] + g2 * d_raw[p + 2];
        }
        __syncthreads();   // last reader of g_raw done -> safe to overwrite

        // ---- prefetch next channel's guidance tile (overlaps with phase B)
        if (c < C - 1)
            load_halo_async(g_raw, gb + (c + 1) * HWp, tile_x, tile_y, tid);

        // ---- phase B: vertical 3-sums, divide, running max over channels
        #pragma unroll
        for (int r = 0; r < 4; ++r) {
            int y = ty + r * 8;                // interior row 0..31
            int p = y * T + tx;                // rs rows y..y+2 = 3x3 window
            float num = rs_p[p] + rs_p[p + T] + rs_p[p + 2 * T];
            float den = rs_g[p] + rs_g[p + T] + rs_g[p + 2 * T];
            maxv[r] = fmaxf(maxv[r], num / den);
        }
        wait_async0();     // next channel's tile fully landed
        __syncthreads();   // rs_* reads done before next phase A overwrites
    }

    // ---- sparse re-insertion + coalesced store
    #pragma unroll
    for (int r = 0; r < 4; ++r) {
        int gi = (tile_y + ty + r * 8) * W + (tile_x + tx);
        float s = spb[gi];
        outb[gi] = (s > 0.f) ? s : maxv[r];
    }
}

extern "C" void kernel_launch(void* const* d_in, const int* in_sizes, int n_in,
                              void* d_out, int out_size, void* d_ws, size_t ws_size,
                              hipStream_t stream) {
    (void)in_sizes; (void)n_in; (void)out_size; (void)ws_size;
    const float* guidance = (const float*)d_in[0];   // [4,8,512,1024]
    const float* blur     = (const float*)d_in[1];   // [4,1,512,1024]
    const float* sparse   = (const float*)d_in[2];   // [4,1,512,1024]
    float* out = (float*)d_out;                      // [4,1,512,1024]
    float* ws  = (float*)d_ws;                       // ping buffer, 8 MB

    const int n  = NB * HWp;
    const int n4 = n / 4;
    cspn_init<<<n4 / NTHREADS, NTHREADS, 0, stream>>>(
        (const float4*)blur, (const float4*)sparse, (float4*)out, n4);

    dim3 grid(W / T, H / T, NB), block(NTHREADS);
    float* bufs[2] = { out, ws };
    for (int s = 0; s < 16; ++s) {
        const float* dsrc = bufs[s & 1];         // step 0 reads d_out (init)
        float*       ddst = bufs[(s + 1) & 1];   // step 15 writes d_out (even count)
        cspn_step<<<grid, block, 0, stream>>>(guidance, sparse, dsrc, ddst);
    }
}